// SingleInputGNN_48352741818715
// MI455X (gfx1250) — compile-verified
//
#include <hip/hip_runtime.h>
#include <math.h>
#include <stdint.h>

// Problem constants (match reference)
#define FF 128   // feature dim
#define HH 128   // hidden dim
#define LL 4     // residual blocks

typedef __attribute__((ext_vector_type(2))) float v2f;
typedef __attribute__((ext_vector_type(8))) float v8f;
typedef __attribute__((ext_vector_type(4))) unsigned int u32x4;
typedef __attribute__((ext_vector_type(8))) unsigned int u32x8;

__device__ __forceinline__ float gelu_exact(float x) {
    return 0.5f * x * (1.0f + erff(x * 0.70710678118654752440f));
}

__device__ __forceinline__ v8f wmma_f32_step(v2f a, v2f b, v8f acc) {
    return __builtin_amdgcn_wmma_f32_16x16x4_f32(
        false, a, false, b, (short)0, acc, false, false);
}

// ---------------------------------------------------------------------------
// Tensor Data Mover: one wave-level DMA of a 2-D f32 tile into LDS.
// Descriptor per CDNA5 ISA ch.8 (D# group0 128b + group1 256b), 2-D form
// (VADDR2/3 NULL). OOB rows (y >= tensor_d1) are zero-filled by the TDM.
// Optional LDS padding interleaves two sources row-by-row (FFN1 concat).
// ---------------------------------------------------------------------------
__device__ __forceinline__ void tdm_load_2d(unsigned lds_addr, const void* gptr,
                                            unsigned tensor_d0, unsigned tensor_d1,
                                            unsigned tile_d0, unsigned tile_d1,
                                            unsigned stride0_elems,
                                            unsigned pad_en, unsigned pad_interval,
                                            unsigned pad_amount) {
    const unsigned long long ga = (unsigned long long)(uintptr_t)gptr;
    u32x4 g0;
    g0[0] = 1u;                                            // count=1 (valid), user mode
    g0[1] = lds_addr;                                      // LDS byte address
    g0[2] = (unsigned)ga;                                  // global_addr[31:0]
    g0[3] = (unsigned)((ga >> 32) & 0x01ffffffull)         // global_addr[56:32]
          | (2u << 30);                                    // type=2 ("image")
    u32x8 g1;
    g1[0] = (2u << 16)                                     // data_size: 4 bytes
          | (pad_en << 20) | (pad_interval << 22) | (pad_amount << 25);
    g1[1] = (tensor_d0 & 0xffffu) << 16;                   // tensor_dim0[15:0] @ bits63:48
    g1[2] = (tensor_d0 >> 16) | ((tensor_d1 & 0xffffu) << 16);
    g1[3] = (tensor_d1 >> 16) | (tile_d0 << 16);           // tile_dim0 @ bits127:112
    g1[4] = tile_d1 & 0xffffu;                             // tile_dim1; tile_dim2=0
    g1[5] = stride0_elems;                                 // tensor_dim0_stride[31:0]
    g1[6] = 0u;                                            // stride0[47:32], stride1[15:0]
    g1[7] = 0u;                                            // stride1[47:16]
    asm volatile("tensor_load_to_lds %0, %1" :: "s"(g0), "s"(g1) : "memory");
}

__device__ __forceinline__ void wait_tensor_all() {
    __builtin_amdgcn_s_wait_tensorcnt(0);
}

// ---------------------------------------------------------------------------
// GEMM core: block = 8 waves covering a 32(M) x 128(N) tile.
// Wave computes 16x32 (two 16x16 WMMA tiles, shared A fragment from LDS).
// A tile staged by the TDM; B pre-transposed so a fragment is one b64.
// ---------------------------------------------------------------------------

// GEMM 1: h = gelu(x @ w_in + b_in); A: Mx128, BT: 128x128
__global__ void gemm_in_kernel(const float* __restrict__ A,
                               const float* __restrict__ BT,
                               const float* __restrict__ bias,
                               float* __restrict__ D, int M) {
    const int K = FF, ldd = HH;
    __shared__ float sA[32 * FF];
    const int m0 = blockIdx.y * 32;

    if ((threadIdx.x >> 5) == 0) {                 // wave 0 drives the DMA
        tdm_load_2d((unsigned)(uintptr_t)sA, A + (size_t)m0 * K,
                    K, (unsigned)(M - m0), K, 32, K, 0, 0, 0);
        wait_tensor_all();
    }
    __syncthreads();

    const int w = threadIdx.x >> 5, lane = threadIdx.x & 31;
    const int half = lane >> 4, r = lane & 15;
    const int mw = m0 + (w >> 2) * 16;
    const int n0 = (blockIdx.x * 4 + (w & 3)) * 32;
    if (mw + 16 <= M) {                            // wave-uniform guard
        const float* sArow = sA + (size_t)((w >> 2) * 16 + r) * K;
        const float* bt0 = BT + (size_t)(n0 + r) * K;
        const float* bt1 = BT + (size_t)(n0 + 16 + r) * K;
        v8f acc0 = {}, acc1 = {};
        for (int k = 0; k < K; k += 4) {
            const v2f a  = *(const v2f*)(sArow + k + 2 * half);
            const v2f b0 = *(const v2f*)(bt0 + k + 2 * half);
            const v2f b1 = *(const v2f*)(bt1 + k + 2 * half);
            acc0 = wmma_f32_step(a, b0, acc0);
            acc1 = wmma_f32_step(a, b1, acc1);
        }
        const int c0 = n0 + r, c1 = n0 + 16 + r;
        const float bb0 = bias[c0], bb1 = bias[c1];
#pragma unroll
        for (int rr = 0; rr < 8; ++rr) {
            const int row = mw + rr + 8 * half;
            D[(size_t)row * ldd + c0] = gelu_exact(acc0[rr] + bb0);
            D[(size_t)row * ldd + c1] = gelu_exact(acc1[rr] + bb1);
        }
    }
}

// GEMM 2 (FFN1): z3 = gelu(concat(z, agg) @ w1 + b1); virtual A: Mx256
// Two padded TDM loads interleave z / agg half-rows directly into sA[32][256].
__global__ void gemm_ffn1_kernel(const float* __restrict__ Z,
                                 const float* __restrict__ AGG,
                                 const float* __restrict__ BT,
                                 const float* __restrict__ bias,
                                 float* __restrict__ D, int M) {
    const int K = 2 * HH, ldd = 2 * HH;
    __shared__ float sA[32 * 2 * HH];
    const int m0 = blockIdx.y * 32;

    if ((threadIdx.x >> 5) == 0) {
        const unsigned base = (unsigned)(uintptr_t)sA;
        // pad_interval=6 -> pad after every 2^6 * 8B = 512B; pad_amount=127 -> 512B
        tdm_load_2d(base,       Z   + (size_t)m0 * HH, HH, (unsigned)(M - m0),
                    HH, 32, HH, 1, 6, 127);
        tdm_load_2d(base + 512, AGG + (size_t)m0 * HH, HH, (unsigned)(M - m0),
                    HH, 32, HH, 1, 6, 127);
        wait_tensor_all();
    }
    __syncthreads();

    const int w = threadIdx.x >> 5, lane = threadIdx.x & 31;
    const int half = lane >> 4, r = lane & 15;
    const int mw = m0 + (w >> 2) * 16;
    const int n0 = (blockIdx.x * 4 + (w & 3)) * 32;
    if (mw + 16 <= M) {
        const float* sArow = sA + (size_t)((w >> 2) * 16 + r) * K;
        const float* bt0 = BT + (size_t)(n0 + r) * K;
        const float* bt1 = BT + (size_t)(n0 + 16 + r) * K;
        v8f acc0 = {}, acc1 = {};
        for (int k = 0; k < K; k += 4) {
            const v2f a  = *(const v2f*)(sArow + k + 2 * half);
            const v2f b0 = *(const v2f*)(bt0 + k + 2 * half);
            const v2f b1 = *(const v2f*)(bt1 + k + 2 * half);
            acc0 = wmma_f32_step(a, b0, acc0);
            acc1 = wmma_f32_step(a, b1, acc1);
        }
        const int c0 = n0 + r, c1 = n0 + 16 + r;
        const float bb0 = bias[c0], bb1 = bias[c1];
#pragma unroll
        for (int rr = 0; rr < 8; ++rr) {
            const int row = mw + rr + 8 * half;
            D[(size_t)row * ldd + c0] = gelu_exact(acc0[rr] + bb0);
            D[(size_t)row * ldd + c1] = gelu_exact(acc1[rr] + bb1);
        }
    }
}

// GEMM 3 (FFN2 + residual): h += z3 @ w2 + b2; A: Mx256, BT: 128x256
__global__ void gemm_ffn2_kernel(const float* __restrict__ A,
                                 const float* __restrict__ BT,
                                 const float* __restrict__ bias,
                                 float* __restrict__ Hbuf, int M) {
    const int K = 2 * HH, ldd = HH;
    __shared__ float sA[32 * 2 * HH];
    const int m0 = blockIdx.y * 32;

    if ((threadIdx.x >> 5) == 0) {
        tdm_load_2d((unsigned)(uintptr_t)sA, A + (size_t)m0 * K,
                    K, (unsigned)(M - m0), K, 32, K, 0, 0, 0);
        wait_tensor_all();
    }
    __syncthreads();

    const int w = threadIdx.x >> 5, lane = threadIdx.x & 31;
    const int half = lane >> 4, r = lane & 15;
    const int mw = m0 + (w >> 2) * 16;
    const int n0 = (blockIdx.x * 4 + (w & 3)) * 32;
    if (mw + 16 <= M) {
        const float* sArow = sA + (size_t)((w >> 2) * 16 + r) * K;
        const float* bt0 = BT + (size_t)(n0 + r) * K;
        const float* bt1 = BT + (size_t)(n0 + 16 + r) * K;
        v8f acc0 = {}, acc1 = {};
        for (int k = 0; k < K; k += 4) {
            const v2f a  = *(const v2f*)(sArow + k + 2 * half);
            const v2f b0 = *(const v2f*)(bt0 + k + 2 * half);
            const v2f b1 = *(const v2f*)(bt1 + k + 2 * half);
            acc0 = wmma_f32_step(a, b0, acc0);
            acc1 = wmma_f32_step(a, b1, acc1);
        }
        const int c0 = n0 + r, c1 = n0 + 16 + r;
        const float bb0 = bias[c0], bb1 = bias[c1];
#pragma unroll
        for (int rr = 0; rr < 8; ++rr) {
            const int row = mw + rr + 8 * half;
            float* p0 = Hbuf + (size_t)row * ldd + c0;
            float* p1 = Hbuf + (size_t)row * ldd + c1;
            *p0 = *p0 + acc0[rr] + bb0;
            *p1 = *p1 + acc1[rr] + bb1;
        }
    }
}

// ---------------------------------------------------------------------------
// LayerNorm: one wave32 per 128-wide row, shuffle reductions
// ---------------------------------------------------------------------------
__global__ void ln_kernel(const float* __restrict__ Hbuf, float* __restrict__ Z,
                          const float* __restrict__ scale,
                          const float* __restrict__ bias, int n) {
    const int row  = (int)((blockIdx.x * (size_t)blockDim.x + threadIdx.x) >> 5);
    const int lane = threadIdx.x & 31;
    if (row >= n) return;
    const float* rp = Hbuf + (size_t)row * HH;
    float v[4], sum = 0.0f;
#pragma unroll
    for (int j = 0; j < 4; ++j) { v[j] = rp[lane + 32 * j]; sum += v[j]; }
#pragma unroll
    for (int off = 16; off > 0; off >>= 1) sum += __shfl_xor(sum, off, 32);
    const float mu = sum * (1.0f / HH);
    float var = 0.0f;
#pragma unroll
    for (int j = 0; j < 4; ++j) { const float d = v[j] - mu; var += d * d; }
#pragma unroll
    for (int off = 16; off > 0; off >>= 1) var += __shfl_xor(var, off, 32);
    const float rstd = rsqrtf(var * (1.0f / HH) + 1e-5f);
#pragma unroll
    for (int j = 0; j < 4; ++j) {
        const int c = lane + 32 * j;
        Z[(size_t)row * HH + c] = (v[j] - mu) * rstd * scale[c] + bias[c];
    }
}

// ---------------------------------------------------------------------------
// CSR build: histogram -> scan -> bucket fill (atomic-free aggregation after)
// ---------------------------------------------------------------------------
__global__ void hist_kernel(const int* __restrict__ edst, int* __restrict__ cnt, int E) {
    const int e = blockIdx.x * blockDim.x + threadIdx.x;
    if (e < E) atomicAdd(&cnt[edst[e]], 1);
}

__global__ void scan_kernel(const int* __restrict__ cnt, int* __restrict__ rowptr, int n) {
    __shared__ int smem[1024];
    __shared__ int carry;
    if (threadIdx.x == 0) { carry = 0; rowptr[0] = 0; }
    __syncthreads();
    for (int base = 0; base < n; base += 1024) {
        const int i = base + (int)threadIdx.x;
        smem[threadIdx.x] = (i < n) ? cnt[i] : 0;
        __syncthreads();
        for (int off = 1; off < 1024; off <<= 1) {
            const int t = (threadIdx.x >= (unsigned)off) ? smem[threadIdx.x - off] : 0;
            __syncthreads();
            smem[threadIdx.x] += t;
            __syncthreads();
        }
        const int incl = smem[threadIdx.x] + carry;
        if (i < n) rowptr[i + 1] = incl;
        __syncthreads();
        if (threadIdx.x == 1023) carry += smem[1023];
        __syncthreads();
    }
}

__global__ void csr_fill_kernel(const int* __restrict__ esrc, const int* __restrict__ edst,
                                int* __restrict__ cursor, int* __restrict__ csr_src, int E) {
    const int e = blockIdx.x * blockDim.x + threadIdx.x;
    if (e < E) {
        const int pos = atomicAdd(&cursor[edst[e]], 1);
        csr_src[pos] = esrc[e];
    }
}

// Gather-mean: one wave per node; lane carries a float4 column slice.
__global__ void gather_kernel(const float* __restrict__ Z,
                              const int* __restrict__ rowptr,
                              const int* __restrict__ csr_src,
                              float* __restrict__ AGG, int n) {
    const int node = (int)((blockIdx.x * (size_t)blockDim.x + threadIdx.x) >> 5);
    const int lane = threadIdx.x & 31;
    if (node >= n) return;
    const int beg = rowptr[node], end = rowptr[node + 1];
    float4 acc = make_float4(0.f, 0.f, 0.f, 0.f);
    for (int e = beg; e < end; ++e) {
        const int s = csr_src[e];
        const float4 v = ((const float4*)(Z + (size_t)s * HH))[lane];
        acc.x += v.x; acc.y += v.y; acc.z += v.z; acc.w += v.w;
    }
    const float inv = 1.0f / fmaxf((float)(end - beg), 1.0f);
    acc.x *= inv; acc.y *= inv; acc.z *= inv; acc.w *= inv;
    ((float4*)(AGG + (size_t)node * HH))[lane] = acc;
}

// Batched transpose: out[b][n][k] = in[b][k][n]  (weights are tiny)
__global__ void transpose_kernel(const float* __restrict__ in, float* __restrict__ out,
                                 int K, int Ncols, int batch) {
    const size_t t = blockIdx.x * (size_t)blockDim.x + threadIdx.x;
    const size_t total = (size_t)batch * K * Ncols;
    if (t >= total) return;
    const int b = (int)(t / ((size_t)K * Ncols));
    const int rem = (int)(t % ((size_t)K * Ncols));
    const int k = rem / Ncols, n = rem % Ncols;
    out[(size_t)b * K * Ncols + (size_t)n * K + k] = in[t];
}

// ---------------------------------------------------------------------------
// Final: out = LN(h) @ w_out + b_out
// ---------------------------------------------------------------------------
__global__ void head_kernel(const float* __restrict__ Hbuf,
                            const float* __restrict__ scale,
                            const float* __restrict__ bias,
                            const float* __restrict__ w_out,
                            const float* __restrict__ b_out,
                            float* __restrict__ out, int n) {
    const int row  = (int)((blockIdx.x * (size_t)blockDim.x + threadIdx.x) >> 5);
    const int lane = threadIdx.x & 31;
    if (row >= n) return;
    const float* rp = Hbuf + (size_t)row * HH;
    float v[4], sum = 0.0f;
#pragma unroll
    for (int j = 0; j < 4; ++j) { v[j] = rp[lane + 32 * j]; sum += v[j]; }
#pragma unroll
    for (int off = 16; off > 0; off >>= 1) sum += __shfl_xor(sum, off, 32);
    const float mu = sum * (1.0f / HH);
    float var = 0.0f;
#pragma unroll
    for (int j = 0; j < 4; ++j) { const float d = v[j] - mu; var += d * d; }
#pragma unroll
    for (int off = 16; off > 0; off >>= 1) var += __shfl_xor(var, off, 32);
    const float rstd = rsqrtf(var * (1.0f / HH) + 1e-5f);
    float dot = 0.0f;
#pragma unroll
    for (int j = 0; j < 4; ++j) {
        const int c = lane + 32 * j;
        dot += ((v[j] - mu) * rstd * scale[c] + bias[c]) * w_out[c];
    }
#pragma unroll
    for (int off = 16; off > 0; off >>= 1) dot += __shfl_xor(dot, off, 32);
    if (lane == 0) out[row] = dot + b_out[0];
}

// ---------------------------------------------------------------------------
extern "C" void kernel_launch(void* const* d_in, const int* in_sizes, int n_in,
                              void* d_out, int out_size, void* d_ws, size_t ws_size,
                              hipStream_t stream) {
    const float* x       = (const float*)d_in[0];
    const int*   esrc    = (const int*)  d_in[1];
    const int*   edst    = (const int*)  d_in[2];
    const float* w_in    = (const float*)d_in[3];
    const float* b_in    = (const float*)d_in[4];
    const float* ln_s    = (const float*)d_in[5];
    const float* ln_b    = (const float*)d_in[6];
    const float* ffn_w1  = (const float*)d_in[7];
    const float* ffn_b1  = (const float*)d_in[8];
    const float* ffn_w2  = (const float*)d_in[9];
    const float* ffn_b2  = (const float*)d_in[10];
    const float* o_ln_s  = (const float*)d_in[11];
    const float* o_ln_b  = (const float*)d_in[12];
    const float* w_out   = (const float*)d_in[13];
    const float* b_out   = (const float*)d_in[14];
    float* out = (float*)d_out;

    const int N = in_sizes[0] / FF;   // 50000
    const int E = in_sizes[1];        // 600000

    // ---- workspace layout ----
    float* hbuf = (float*)d_ws;                        // N*H
    float* zbuf = hbuf + (size_t)N * HH;               // N*H
    float* agg  = zbuf + (size_t)N * HH;               // N*H (already mean)
    float* z3   = agg  + (size_t)N * HH;               // N*2H
    float* winT = z3   + (size_t)N * 2 * HH;           // 128*128
    float* w1T  = winT + (size_t)FF * HH;              // 4*256*256
    float* w2T  = w1T  + (size_t)LL * 2 * HH * 2 * HH; // 4*128*256
    int*   cnt     = (int*)(w2T + (size_t)LL * 2 * HH * HH); // N
    int*   rowptr  = cnt + N;                          // N+1
    int*   cursor  = rowptr + (N + 1);                 // N
    int*   csr_src = cursor + N;                       // E

    const int mt32 = (N + 31) / 32;

    // ---- CSR build (per call; edge structure is an input) ----
    hipMemsetAsync(cnt, 0, (size_t)N * sizeof(int), stream);
    hist_kernel<<<(E + 255) / 256, 256, 0, stream>>>(edst, cnt, E);
    scan_kernel<<<1, 1024, 0, stream>>>(cnt, rowptr, N);
    hipMemcpyAsync(cursor, rowptr, (size_t)N * sizeof(int),
                   hipMemcpyDeviceToDevice, stream);
    csr_fill_kernel<<<(E + 255) / 256, 256, 0, stream>>>(esrc, edst, cursor, csr_src, E);

    // ---- weight pre-transpose (contiguous b64 B-fragments) ----
    transpose_kernel<<<(FF * HH + 255) / 256, 256, 0, stream>>>(w_in, winT, FF, HH, 1);
    transpose_kernel<<<(LL * 2 * HH * 2 * HH + 255) / 256, 256, 0, stream>>>(
        ffn_w1, w1T, 2 * HH, 2 * HH, LL);
    transpose_kernel<<<(LL * 2 * HH * HH + 255) / 256, 256, 0, stream>>>(
        ffn_w2, w2T, 2 * HH, HH, LL);

    // ---- h = gelu(x @ w_in + b_in) ----
    gemm_in_kernel<<<dim3(1, mt32), 256, 0, stream>>>(x, winT, b_in, hbuf, N);

    for (int i = 0; i < LL; ++i) {
        ln_kernel<<<(N + 7) / 8, 256, 0, stream>>>(hbuf, zbuf,
                                                   ln_s + (size_t)i * HH,
                                                   ln_b + (size_t)i * HH, N);
        gather_kernel<<<(N + 7) / 8, 256, 0, stream>>>(zbuf, rowptr, csr_src, agg, N);
        gemm_ffn1_kernel<<<dim3(2, mt32), 256, 0, stream>>>(
            zbuf, agg,
            w1T + (size_t)i * 2 * HH * 2 * HH,
            ffn_b1 + (size_t)i * 2 * HH, z3, N);
        gemm_ffn2_kernel<<<dim3(1, mt32), 256, 0, stream>>>(
            z3,
            w2T + (size_t)i * 2 * HH * HH,
            ffn_b2 + (size_t)i * HH, hbuf, N);
    }

    head_kernel<<<(N + 7) / 8, 256, 0, stream>>>(hbuf, o_ln_s, o_ln_b,
                                                 w_out, b_out, out, N);
}